// Model_48996986913171
// MI455X (gfx1250) — compile-verified
//
#include <hip/hip_runtime.h>
#include <math.h>

typedef __attribute__((ext_vector_type(2))) float v2f;
typedef __attribute__((ext_vector_type(8))) float v8f;

// Problem constants (match reference)
constexpr int HOPS = 3;
constexpr int V    = 32000;  // NWORDS
constexpr int E    = 256;    // EMB
constexpr int Mm   = 512;    // memory slots
constexpr int Bb   = 64;     // batch
constexpr int Tr   = 4;      // tokens per memory line
constexpr int Tt   = 32;     // decode steps
constexpr int K2   = 2 * E;  // 512, vocab GEMM K

// ---------------------------------------------------------------------------
// Encoder: scores with on-the-fly embedding gather.  One wave per (b,m).
// s[b,m] = sum_tr dot(table[ctx[m,b,tr]], q[b])
// ---------------------------------------------------------------------------
__global__ void enc_scores_kernel(const float* __restrict__ table,
                                  const int* __restrict__ ctx,
                                  const float* __restrict__ q,
                                  float* __restrict__ s) {
  const int lane = threadIdx.x & 31;
  const int wave = threadIdx.x >> 5;
  const int b = blockIdx.y;
  const int m = blockIdx.x * 8 + wave;
  const int* cx = ctx + ((size_t)m * Bb + b) * Tr;
  const float* qv = q + (size_t)b * E;
  float acc = 0.f;
  #pragma unroll
  for (int tr = 0; tr < Tr; ++tr) {
    const float* row = table + (size_t)cx[tr] * E;
    #pragma unroll 4
    for (int e = lane; e < E; e += 32) acc += row[e] * qv[e];
  }
  #pragma unroll
  for (int off = 16; off; off >>= 1) acc += __shfl_xor(acc, off, 32);
  if (lane == 0) s[(size_t)b * Mm + m] = acc;
}

// ---------------------------------------------------------------------------
// Decoder attention scores from precomputed mems.  One wave per (b,m).
// Optionally also writes raw scores to pout (all_ptr for hop 2).
// ---------------------------------------------------------------------------
__global__ void scores_kernel(const float* __restrict__ mem,
                              const float* __restrict__ q,
                              float* __restrict__ s,
                              float* __restrict__ pout) {
  const int lane = threadIdx.x & 31;
  const int wave = threadIdx.x >> 5;
  const int b = blockIdx.y;
  const int m = blockIdx.x * 8 + wave;
  const float* mv = mem + ((size_t)b * Mm + m) * E;
  const float* qv = q + (size_t)b * E;
  float acc = 0.f;
  #pragma unroll 4
  for (int e = lane; e < E; e += 32) acc += mv[e] * qv[e];
  #pragma unroll
  for (int off = 16; off; off >>= 1) acc += __shfl_xor(acc, off, 32);
  if (lane == 0) {
    s[(size_t)b * Mm + m] = acc;
    if (pout) pout[(size_t)b * Mm + m] = acc;
  }
}

// ---------------------------------------------------------------------------
// Softmax over M=512 per batch row.  One block of 256 threads per b.
// ---------------------------------------------------------------------------
__global__ void softmax_kernel(const float* __restrict__ s,
                               float* __restrict__ attn) {
  const int b = blockIdx.x;
  __shared__ float red[256];
  const float* sb = s + (size_t)b * Mm;
  float* ab = attn + (size_t)b * Mm;
  float vmax = -3.4e38f;
  for (int m = threadIdx.x; m < Mm; m += 256) vmax = fmaxf(vmax, sb[m]);
  red[threadIdx.x] = vmax; __syncthreads();
  for (int st = 128; st > 0; st >>= 1) {
    if (threadIdx.x < st) red[threadIdx.x] = fmaxf(red[threadIdx.x], red[threadIdx.x + st]);
    __syncthreads();
  }
  vmax = red[0]; __syncthreads();
  float sum = 0.f;
  for (int m = threadIdx.x; m < Mm; m += 256) {
    float e = __expf(sb[m] - vmax);
    ab[m] = e; sum += e;
  }
  red[threadIdx.x] = sum; __syncthreads();
  for (int st = 128; st > 0; st >>= 1) {
    if (threadIdx.x < st) red[threadIdx.x] += red[threadIdx.x + st];
    __syncthreads();
  }
  const float inv = 1.0f / red[0];
  for (int m = threadIdx.x; m < Mm; m += 256) ab[m] *= inv;
}

// ---------------------------------------------------------------------------
// Encoder weighted sum with gather: q[b,e] += sum_m attn[b,m]*sum_tr C[ctx][e]
// One block (256 threads = e) per b.
// ---------------------------------------------------------------------------
__global__ void enc_weighted_kernel(const float* __restrict__ table,
                                    const int* __restrict__ ctx,
                                    const float* __restrict__ attn,
                                    float* __restrict__ q) {
  const int b = blockIdx.x;
  const int e = threadIdx.x;
  __shared__ float a[Mm];
  for (int m = threadIdx.x; m < Mm; m += 256) a[m] = attn[(size_t)b * Mm + m];
  __syncthreads();
  float acc = 0.f;
  for (int m = 0; m < Mm; ++m) {
    const int* cx = ctx + ((size_t)m * Bb + b) * Tr;
    float c = 0.f;
    #pragma unroll
    for (int tr = 0; tr < Tr; ++tr) c += table[(size_t)cx[tr] * E + e];
    acc += a[m] * c;
  }
  q[(size_t)b * E + e] += acc;
}

// ---------------------------------------------------------------------------
// Decoder weighted sum: qc[b,e] += sum_m attn[b,m]*mem[b,m,e]; hop0 also
// writes o0 into the second half of X (vocab GEMM input).
// ---------------------------------------------------------------------------
__global__ void weighted_kernel(const float* __restrict__ mem,
                                const float* __restrict__ attn,
                                float* __restrict__ qc,
                                float* __restrict__ Xo) {
  const int b = blockIdx.x;
  const int e = threadIdx.x;
  __shared__ float a[Mm];
  for (int m = threadIdx.x; m < Mm; m += 256) a[m] = attn[(size_t)b * Mm + m];
  __syncthreads();
  const float* mp = mem + (size_t)b * Mm * E + e;
  float acc = 0.f;
  #pragma unroll 4
  for (int m = 0; m < Mm; ++m) acc += a[m] * mp[(size_t)m * E];
  qc[(size_t)b * E + e] += acc;
  if (Xo) Xo[(size_t)b * K2 + E + e] = acc;
}

// ---------------------------------------------------------------------------
// Precompute decoder memories: mems[hp,b,m,e] = sum_tr dec_C[hp][ctx[m,b,tr]][e]
// grid (Mm, Bb, HOPS+1), 256 threads = e
// ---------------------------------------------------------------------------
__global__ void mems_kernel(const float* __restrict__ dec_C,
                            const int* __restrict__ ctx,
                            float* __restrict__ mems) {
  const int m = blockIdx.x, b = blockIdx.y, hp = blockIdx.z;
  const int e = threadIdx.x;
  const int* cx = ctx + ((size_t)m * Bb + b) * Tr;
  const float* tab = dec_C + (size_t)hp * V * E;
  float acc = 0.f;
  #pragma unroll
  for (int tr = 0; tr < Tr; ++tr) acc += tab[(size_t)cx[tr] * E + e];
  mems[(((size_t)hp * Bb + b) * Mm + m) * E + e] = acc;
}

// ---------------------------------------------------------------------------
// GRU cell, one block (256 threads = e) per b.  Also seeds qc and X[:, :E].
// ---------------------------------------------------------------------------
__global__ void gru_kernel(const float* __restrict__ dec_C0,
                           const int* __restrict__ targets, int t,
                           const float* __restrict__ W_ih,
                           const float* __restrict__ W_hh,
                           const float* __restrict__ b_ih,
                           const float* __restrict__ b_hh,
                           const float* __restrict__ h_in,
                           float* __restrict__ h_out,
                           float* __restrict__ qc,
                           float* __restrict__ X) {
  const int b = blockIdx.x;
  const int e = threadIdx.x;
  __shared__ float xs[E], hs[E];
  const int y = (t == 0) ? 2 : targets[(size_t)(t - 1) * Bb + b];
  xs[e] = dec_C0[(size_t)y * E + e];
  hs[e] = h_in[(size_t)b * E + e];
  __syncthreads();
  float gi[3], gh[3];
  #pragma unroll
  for (int g = 0; g < 3; ++g) {
    const float* wi = W_ih + (size_t)(g * E + e) * E;
    const float* wh = W_hh + (size_t)(g * E + e) * E;
    float si = b_ih[g * E + e], sh = b_hh[g * E + e];
    #pragma unroll 4
    for (int k = 0; k < E; ++k) { si += wi[k] * xs[k]; sh += wh[k] * hs[k]; }
    gi[g] = si; gh[g] = sh;
  }
  const float r = 1.0f / (1.0f + __expf(-(gi[0] + gh[0])));
  const float z = 1.0f / (1.0f + __expf(-(gi[1] + gh[1])));
  const float n = tanhf(gi[2] + r * gh[2]);
  const float hn = (1.0f - z) * n + z * hs[e];
  h_out[(size_t)b * E + e] = hn;
  qc[(size_t)b * E + e]    = hn;
  X[(size_t)b * K2 + e]    = hn;   // q0 half of vocab GEMM input
}

// ---------------------------------------------------------------------------
// Vocab projection: out[b,v] = X[b,:] . W_vocab[v,:] + b_vocab[v]
// fp32 WMMA 16x16x4.  Block: 256 threads = 8 waves; each wave owns one 16x16
// tile (16 batch rows x 16 vocab words); A tile staged in padded LDS.
// 4 independent accumulator chains over K to hide XDL latency.
// ---------------------------------------------------------------------------
#define XS_STRIDE 516  // 512 + 4 pad -> rows land on distinct LDS banks
__global__ void vocab_wmma_kernel(const float* __restrict__ X,
                                  const float* __restrict__ Wv,
                                  const float* __restrict__ bv,
                                  float* __restrict__ out) {
  __shared__ float xs[16 * XS_STRIDE];
  const int i0 = blockIdx.y * 16;  // batch tile
  // cooperative load of A tile [16 x K2]
  for (int idx = threadIdx.x; idx < 16 * K2; idx += 256) {
    const int r = idx >> 9;        // / 512
    const int c = idx & (K2 - 1);
    xs[r * XS_STRIDE + c] = X[(size_t)(i0 + r) * K2 + c];
  }
  __syncthreads();

  const int lane = threadIdx.x & 31;
  const int wave = threadIdx.x >> 5;
  const int j0 = (blockIdx.x * 8 + wave) * 16;  // vocab tile
  const int half = lane >> 4;                    // K sub-pair select
  const int l = lane & 15;                       // M for A, N for B/C/D
  const int koff = half * 2;
  const float* xrow = xs + l * XS_STRIDE;
  const float* wrow = Wv + (size_t)(j0 + l) * K2;

  v8f acc0 = {}, acc1 = {}, acc2 = {}, acc3 = {};
  for (int k0 = 0; k0 < K2; k0 += 16) {
    v2f a0 = *(const v2f*)(xrow + k0 + 0  + koff);
    v2f b0 = *(const v2f*)(wrow + k0 + 0  + koff);
    v2f a1 = *(const v2f*)(xrow + k0 + 4  + koff);
    v2f b1 = *(const v2f*)(wrow + k0 + 4  + koff);
    v2f a2 = *(const v2f*)(xrow + k0 + 8  + koff);
    v2f b2 = *(const v2f*)(wrow + k0 + 8  + koff);
    v2f a3 = *(const v2f*)(xrow + k0 + 12 + koff);
    v2f b3 = *(const v2f*)(wrow + k0 + 12 + koff);
    acc0 = __builtin_amdgcn_wmma_f32_16x16x4_f32(false, a0, false, b0, (short)0, acc0, false, false);
    acc1 = __builtin_amdgcn_wmma_f32_16x16x4_f32(false, a1, false, b1, (short)0, acc1, false, false);
    acc2 = __builtin_amdgcn_wmma_f32_16x16x4_f32(false, a2, false, b2, (short)0, acc2, false, false);
    acc3 = __builtin_amdgcn_wmma_f32_16x16x4_f32(false, a3, false, b3, (short)0, acc3, false, false);
  }
  const float bias = bv[j0 + l];
  #pragma unroll
  for (int r = 0; r < 8; ++r) {
    const int row = i0 + r + half * 8;      // C/D layout: lanes 16-31 hold M+8
    out[(size_t)row * V + (j0 + l)] = (acc0[r] + acc1[r]) + (acc2[r] + acc3[r]) + bias;
  }
}

// ---------------------------------------------------------------------------
extern "C" void kernel_launch(void* const* d_in, const int* in_sizes, int n_in,
                              void* d_out, int out_size, void* d_ws, size_t ws_size,
                              hipStream_t stream) {
  const int*   ctx     = (const int*)d_in[0];    // [M,B,Tr]
  const int*   targets = (const int*)d_in[1];    // [T,B]
  const float* enc_C   = (const float*)d_in[2];  // [4,V,E]
  const float* dec_C   = (const float*)d_in[3];  // [4,V,E]
  const float* W_ih    = (const float*)d_in[4];  // [3E,E]
  const float* W_hh    = (const float*)d_in[5];
  const float* b_ih    = (const float*)d_in[6];
  const float* b_hh    = (const float*)d_in[7];
  const float* Wv      = (const float*)d_in[8];  // [V,2E]
  const float* bv      = (const float*)d_in[9];  // [V]

  // workspace layout (floats): mems is the big one (128 MB)
  float* mems = (float*)d_ws;                        // [4,B,M,E]
  float* hA   = mems + (size_t)(HOPS + 1) * Bb * Mm * E;
  float* hB   = hA + (size_t)Bb * E;
  float* qc   = hB + (size_t)Bb * E;
  float* X    = qc + (size_t)Bb * E;                 // [B, 2E]
  float* s    = X  + (size_t)Bb * K2;                // [B, M]
  float* attn = s  + (size_t)Bb * Mm;                // [B, M]

  float* out_vocab = (float*)d_out;                          // [T,B,V]
  float* out_ptr   = out_vocab + (size_t)Tt * Bb * V;        // [T,B,M]
  float* out_hfin  = out_ptr   + (size_t)Tt * Bb * Mm;       // [B,E]

  const dim3 blk256(256);
  const dim3 grid_sc(Mm / 8, Bb);       // scores: 8 waves/block, wave per (b,m)
  const dim3 grid_voc(V / (16 * 8), Bb / 16);  // (250, 4)

  // ---- Encoder: q starts at zero, 3 hops of gather-attention ----
  hipMemsetAsync(hA, 0, (size_t)Bb * E * sizeof(float), stream);
  for (int hp = 0; hp < HOPS; ++hp) {
    enc_scores_kernel<<<grid_sc, blk256, 0, stream>>>(
        enc_C + (size_t)hp * V * E, ctx, hA, s);
    softmax_kernel<<<Bb, blk256, 0, stream>>>(s, attn);
    enc_weighted_kernel<<<Bb, blk256, 0, stream>>>(
        enc_C + (size_t)(hp + 1) * V * E, ctx, attn, hA);
  }
  // hA now holds h0

  // ---- Precompute decoder memories ----
  mems_kernel<<<dim3(Mm, Bb, HOPS + 1), blk256, 0, stream>>>(dec_C, ctx, mems);

  // ---- Decode loop (teacher forcing) ----
  for (int t = 0; t < Tt; ++t) {
    const float* h_in = (t & 1) ? hB : hA;
    float*       h_out = (t & 1) ? hA : hB;
    gru_kernel<<<Bb, blk256, 0, stream>>>(dec_C /*hp0 table*/, targets, t,
                                          W_ih, W_hh, b_ih, b_hh,
                                          h_in, h_out, qc, X);
    for (int hp = 0; hp < HOPS; ++hp) {
      scores_kernel<<<grid_sc, blk256, 0, stream>>>(
          mems + (size_t)hp * Bb * Mm * E, qc, s,
          (hp == HOPS - 1) ? (out_ptr + (size_t)t * Bb * Mm) : nullptr);
      softmax_kernel<<<Bb, blk256, 0, stream>>>(s, attn);
      weighted_kernel<<<Bb, blk256, 0, stream>>>(
          mems + (size_t)(hp + 1) * Bb * Mm * E, attn, qc,
          (hp == 0) ? X : nullptr);
      if (hp == 0) {
        vocab_wmma_kernel<<<grid_voc, blk256, 0, stream>>>(
            X, Wv, bv, out_vocab + (size_t)t * Bb * V);
      }
    }
  }

  // ---- Final hidden state: t=31 wrote into hA ----
  hipMemcpyAsync(out_hfin, hA, (size_t)Bb * E * sizeof(float),
                 hipMemcpyDeviceToDevice, stream);
}